// LFLF_GAT_39814346834050
// MI455X (gfx1250) — compile-verified
//
#include <hip/hip_runtime.h>
#include <math.h>

// ---- problem constants (match reference) ----
#define N_NODES 50000
#define N_EDGES 800000
#define IN_C    128
#define HID     128
#define CLS     64
#define NHEAD   2
#define ATT_H   64
#define NEG_SLOPE 0.2f

typedef __attribute__((ext_vector_type(2))) float v2f;
typedef __attribute__((ext_vector_type(8))) float v8f;

// ---------------- float atomic max via int/uint trick ----------------
__device__ __forceinline__ void atomicMaxF(float* addr, float v) {
    if (v >= 0.0f)
        atomicMax((int*)addr, __float_as_int(v));
    else
        atomicMin((unsigned int*)addr, __float_as_uint(v));
}

// ---------------- simple fill ----------------
__global__ void fill_kernel(float* __restrict__ p, float v, int n) {
    int i = blockIdx.x * blockDim.x + threadIdx.x;
    if (i < n) p[i] = v;
}

// ---------------- fold attention vector: v[h,k] = sum_j W[k, h*HID+j] * att[h,j] ----------------
__global__ void fold_kernel(const float* __restrict__ W, const float* __restrict__ att,
                            float* __restrict__ out, int in_c) {
    int idx = threadIdx.x;                 // single block, 256 threads
    if (idx >= in_c * NHEAD) return;
    int h = idx / in_c, k = idx % in_c;
    float s = 0.f;
    for (int j = 0; j < HID; ++j)
        s += W[k * (NHEAD * HID) + h * HID + j] * att[h * HID + j];
    out[h * in_c + k] = s;
}

// c_edge[h] = sum_j Wedge[h*HID+j] * att_edge[h,j]   (edge_dim == 1)
__global__ void edge_const_kernel(const float* __restrict__ Wedge,
                                  const float* __restrict__ att_edge,
                                  float* __restrict__ c_edge) {
    int h = threadIdx.x;
    if (h < NHEAD) {
        float s = 0.f;
        for (int j = 0; j < HID; ++j) s += Wedge[h * HID + j] * att_edge[h * HID + j];
        c_edge[h] = s;
    }
}

// ---------------- per-node attention logits via folded vectors (one x-row read for 4 dots) ----
__global__ void logits_kernel(const float* __restrict__ X, const float* __restrict__ vsrc,
                              const float* __restrict__ vdst, float* __restrict__ a_src,
                              float* __restrict__ a_dst, int n, int in_c) {
    int node = blockIdx.x * blockDim.x + threadIdx.x;
    if (node >= n) return;
    const float* xr = X + (size_t)node * in_c;
    float s0 = 0.f, s1 = 0.f, d0 = 0.f, d1 = 0.f;
    for (int k = 0; k < in_c; ++k) {
        float xv = xr[k];
        s0 += xv * vsrc[k];         s1 += xv * vsrc[in_c + k];
        d0 += xv * vdst[k];         d1 += xv * vdst[in_c + k];
    }
    a_src[node * 2] = s0;  a_src[node * 2 + 1] = s1;
    a_dst[node * 2] = d0;  a_dst[node * 2 + 1] = d1;
}

// ---------------- edge pass 1: alpha = lrelu(logits) ; segment max over dst ----------------
__global__ void edge_alpha_kernel(const int* __restrict__ ei,
        const float* __restrict__ asa, const float* __restrict__ ada,
        const float* __restrict__ asl, const float* __restrict__ adl,
        const float* __restrict__ ew, const float* __restrict__ c_edge,
        float* __restrict__ alpha_a, float* __restrict__ alpha_l,
        float* __restrict__ amax_a, float* __restrict__ amax_l) {
    int e = blockIdx.x * blockDim.x + threadIdx.x;
    if (e >= N_EDGES) return;
    int s = ei[e], d = ei[N_EDGES + e];
    float w = ew[e];
#pragma unroll
    for (int h = 0; h < NHEAD; ++h) {
        float va = asa[s * 2 + h] + ada[d * 2 + h];
        va = va >= 0.f ? va : NEG_SLOPE * va;
        alpha_a[e * 2 + h] = va;
        atomicMaxF(&amax_a[d * 2 + h], va);
        float vl = asl[s * 2 + h] + adl[d * 2 + h] + w * c_edge[h];
        vl = vl >= 0.f ? vl : NEG_SLOPE * vl;
        alpha_l[e * 2 + h] = vl;
        atomicMaxF(&amax_l[d * 2 + h], vl);
    }
}

// ---------------- edge pass 2: ex = exp(alpha - amax[dst]); denom += ex; agg += ex * feat[src]
// 64-thread block per edge; thread t covers features 4t..4t+3 (head h = t>>5) for both convs,
// using b128 vector loads; atomic f32 adds are the inherent scatter cost.
__global__ void edge_agg_kernel(const int* __restrict__ ei,
        const float* __restrict__ alpha_a, const float* __restrict__ alpha_l,
        const float* __restrict__ amax_a, const float* __restrict__ amax_l,
        const float* __restrict__ xs, const float* __restrict__ ys,
        float* __restrict__ den_a, float* __restrict__ den_l,
        float* __restrict__ agg_a, float* __restrict__ agg_l) {
    __shared__ int s_src, s_dst;
    __shared__ float s_ex[4];
    int e = blockIdx.x;
    int t = threadIdx.x;                  // 0..63
    if (t == 0) { s_src = ei[e]; s_dst = ei[N_EDGES + e]; }
    __syncthreads();
    if (t < 4) {
        int conv = t >> 1, h = t & 1;
        const float* al = conv ? alpha_l : alpha_a;
        const float* am = conv ? amax_l  : amax_a;
        float exv = expf(al[e * 2 + h] - am[s_dst * 2 + h]);
        s_ex[t] = exv;
        float* den = conv ? den_l : den_a;
        atomicAdd(&den[s_dst * 2 + h], exv);
    }
    __syncthreads();
    int h = t >> 5;
    float ea = s_ex[h];
    float el = s_ex[2 + h];
    float4 xa = ((const float4*)(xs + (size_t)s_src * 256))[t];
    float4 xl = ((const float4*)(ys + (size_t)s_src * 256))[t];
    float* da = agg_a + (size_t)s_dst * 256 + 4 * t;
    float* dl = agg_l + (size_t)s_dst * 256 + 4 * t;
    atomicAdd(da + 0, ea * xa.x); atomicAdd(da + 1, ea * xa.y);
    atomicAdd(da + 2, ea * xa.z); atomicAdd(da + 3, ea * xa.w);
    atomicAdd(dl + 0, el * xl.x); atomicAdd(dl + 1, el * xl.y);
    atomicAdd(dl + 2, el * xl.z); atomicAdd(dl + 3, el * xl.w);
}

// ---------------- normalize + head-mean + bias ----------------
__global__ void finalize_emb_kernel(const float* __restrict__ agg_a, const float* __restrict__ den_a,
                                    const float* __restrict__ bias_a, float* __restrict__ emb_a,
                                    const float* __restrict__ agg_l, const float* __restrict__ den_l,
                                    const float* __restrict__ bias_l, float* __restrict__ emb_l) {
    int idx = blockIdx.x * blockDim.x + threadIdx.x;
    if (idx >= N_NODES * HID) return;
    int n = idx >> 7, k = idx & 127;
    float da0 = den_a[n * 2] + 1e-16f, da1 = den_a[n * 2 + 1] + 1e-16f;
    emb_a[idx] = 0.5f * (agg_a[(size_t)n * 256 + k] / da0 + agg_a[(size_t)n * 256 + 128 + k] / da1) + bias_a[k];
    float dl0 = den_l[n * 2] + 1e-16f, dl1 = den_l[n * 2 + 1] + 1e-16f;
    emb_l[idx] = 0.5f * (agg_l[(size_t)n * 256 + k] / dl0 + agg_l[(size_t)n * 256 + 128 + k] / dl1) + bias_l[k];
}

// ---------------- fp32 WMMA GEMM: C[M,Ncol] = A[M,K] @ B[K,Ncol] (+bias, epilogue) ----------------
// One wave computes TWO vertically adjacent 16x16 tiles sharing one B fragment per k-step:
// 2 wmma per {2x b64 A-load + 2x b32 B-load}. Pointers strength-reduced (no per-iter 64b muls).
// A layout per ISA (32-bit A 16x4): lane = (K>=2)*16 + M; vgpr0=K0, vgpr1=K0+1. B mirrored; C per §7.12.2.
__device__ __forceinline__ float epi_apply(float v, float bv, int epi) {
    if (epi == 1)      return tanhf(v + bv);
    else if (epi == 2) return 1.f / (1.f + expf(-(v + bv)));
    return v;
}

__global__ void wmma_gemm_kernel(const float* __restrict__ A, const float* __restrict__ B,
                                 float* __restrict__ C, int M, int K, int Ncol,
                                 const float* __restrict__ bias, int epi) {
    int wave  = (blockIdx.x * blockDim.x + threadIdx.x) >> 5;
    int lane  = threadIdx.x & 31;
    int tilesN = Ncol >> 4;
    int tilesM = M >> 4;
    int mPairs = (tilesM + 1) >> 1;
    if (wave >= mPairs * tilesN) return;       // wave-uniform: EXEC stays all-1s for WMMA
    int pm  = wave / tilesN;
    int tn  = wave % tilesN;
    int tm0 = pm * 2;
    bool has2 = (tm0 + 1) < tilesM;            // wave-uniform remainder handling
    int half = lane >> 4;
    int lm   = lane & 15;

    const float* ap0 = A + (size_t)(tm0 * 16 + lm) * K + half * 2;
    const float* ap1 = ap0 + (size_t)16 * K;
    const float* bp  = B + (size_t)(half * 2) * Ncol + tn * 16 + lm;
    const size_t bstep = (size_t)Ncol * 4;

    v8f acc0 = {}, acc1 = {};
    if (has2) {
#pragma unroll 4
        for (int k = 0; k < K; k += 4) {
            v2f b; b.x = bp[0]; b.y = bp[Ncol];
            v2f a0 = *(const v2f*)ap0;
            v2f a1 = *(const v2f*)ap1;
            acc0 = __builtin_amdgcn_wmma_f32_16x16x4_f32(false, a0, false, b, (short)0, acc0, false, false);
            acc1 = __builtin_amdgcn_wmma_f32_16x16x4_f32(false, a1, false, b, (short)0, acc1, false, false);
            ap0 += 4; ap1 += 4; bp += bstep;
        }
    } else {
#pragma unroll 4
        for (int k = 0; k < K; k += 4) {
            v2f b; b.x = bp[0]; b.y = bp[Ncol];
            v2f a0 = *(const v2f*)ap0;
            acc0 = __builtin_amdgcn_wmma_f32_16x16x4_f32(false, a0, false, b, (short)0, acc0, false, false);
            ap0 += 4; bp += bstep;
        }
    }

    int n = tn * 16 + lm;
    float bv = bias ? bias[n] : 0.f;
    {
        int mBase = tm0 * 16 + half * 8;
#pragma unroll
        for (int d = 0; d < 8; ++d)
            C[(size_t)(mBase + d) * Ncol + n] = epi_apply(acc0[d], bv, epi);
    }
    if (has2) {
        int mBase = (tm0 + 1) * 16 + half * 8;
#pragma unroll
        for (int d = 0; d < 8; ++d)
            C[(size_t)(mBase + d) * Ncol + n] = epi_apply(acc1[d], bv, epi);
    }
}

// ---------------- semantic attention: beta = softmax over the two channels ----------------
__global__ void beta_kernel(const float* __restrict__ t_a, const float* __restrict__ t_l,
                            const float* __restrict__ W2, float* __restrict__ beta) {
    int n = blockIdx.x * blockDim.x + threadIdx.x;
    if (n >= N_NODES) return;
    float wa = 0.f, wl = 0.f;
    for (int j = 0; j < ATT_H; ++j) {
        float w2 = W2[j];
        wa += t_a[(size_t)n * ATT_H + j] * w2;
        wl += t_l[(size_t)n * ATT_H + j] * w2;
    }
    float m = fmaxf(wa, wl);
    float ea = expf(wa - m), el = expf(wl - m);
    float inv = 1.f / (ea + el);
    beta[n * 2]     = ea * inv;
    beta[n * 2 + 1] = el * inv;
}

__global__ void combine_kernel(const float* __restrict__ emb_a, const float* __restrict__ emb_l,
                               const float* __restrict__ beta, float* __restrict__ xout, int relu) {
    int idx = blockIdx.x * blockDim.x + threadIdx.x;
    if (idx >= N_NODES * HID) return;
    int n = idx >> 7;
    float v = beta[n * 2] * emb_a[idx] + beta[n * 2 + 1] * emb_l[idx];
    if (relu) v = fmaxf(v, 0.f);
    xout[idx] = v;
}

// =================================================================================
extern "C" void kernel_launch(void* const* d_in, const int* in_sizes, int n_in,
                              void* d_out, int out_size, void* d_ws, size_t ws_size,
                              hipStream_t stream) {
    (void)in_sizes; (void)n_in; (void)out_size; (void)ws_size;

    // setup_inputs() dict order (depth-first, insertion order):
    // 0:x 1:y 2:edge_index 3:edge_weights
    // 4..8  convs[0]: Wsrc,Wdst,att_src,att_dst,bias
    // 9..13 convs[1]: Wsrc,Wdst,att_src,att_dst,bias
    // 14..20 label:   Wsrc,Wdst,att_src,att_dst,bias,Wedge,att_edge
    // 21 mlp_W  22 mlp_b  23 att_W1  24 att_b1  25 att_W2
    const float* x0 = (const float*)d_in[0];
    const float* y0 = (const float*)d_in[1];
    const int*   ei = (const int*)d_in[2];
    const float* ew = (const float*)d_in[3];
    const float* cW[2][5];
    for (int i = 0; i < 2; ++i)
        for (int j = 0; j < 5; ++j) cW[i][j] = (const float*)d_in[4 + i * 5 + j];
    const float* lb[7];
    for (int j = 0; j < 7; ++j) lb[j] = (const float*)d_in[14 + j];
    const float* mlp_W  = (const float*)d_in[21];
    const float* mlp_b  = (const float*)d_in[22];
    const float* att_W1 = (const float*)d_in[23];
    const float* att_b1 = (const float*)d_in[24];
    const float* att_W2 = (const float*)d_in[25];

    // ---- workspace layout ----
    float* ws = (float*)d_ws;
    size_t off = 0;
    auto alloc = [&](size_t nfl) { float* p = ws + off; off += nfl; return p; };
    const size_t N = N_NODES, E = N_EDGES;
    float* xs      = alloc(N * 256);
    float* ys      = alloc(N * 256);
    float* asa     = alloc(N * 2);
    float* ada     = alloc(N * 2);
    float* asl     = alloc(N * 2);
    float* adl     = alloc(N * 2);
    float* alpha_a = alloc(E * 2);
    float* alpha_l = alloc(E * 2);
    float* amax_a  = alloc(N * 2);     // amax_a/amax_l contiguous for one fill
    float* amax_l  = alloc(N * 2);
    float* den_a   = alloc(N * 2);     // den_a/den_l contiguous
    float* den_l   = alloc(N * 2);
    float* agg_a   = alloc(N * 256);   // agg_a/agg_l contiguous
    float* agg_l   = alloc(N * 256);
    float* emb_a   = alloc(N * HID);
    float* emb_l   = alloc(N * HID);
    float* t_a     = alloc(N * ATT_H);
    float* t_l     = alloc(N * ATT_H);
    float* beta    = alloc(N * 2);
    float* x_buf   = alloc(N * HID);
    float* y_buf   = alloc(N * CLS);
    float* vsrc_adj = alloc(NHEAD * IN_C);
    float* vdst_adj = alloc(NHEAD * IN_C);
    float* vsrc_lab = alloc(NHEAD * CLS);
    float* vdst_lab = alloc(NHEAD * CLS);
    float* c_edge   = alloc(NHEAD);

    auto cdiv = [](size_t a, size_t b) { return (int)((a + b - 1) / b); };
    auto gemm = [&](const float* A, const float* B, float* C, int M, int K, int Ncol,
                    const float* bias, int epi) {
        int mPairs = (M / 16 + 1) / 2;
        int tiles  = mPairs * (Ncol / 16);
        wmma_gemm_kernel<<<(tiles + 7) / 8, 256, 0, stream>>>(A, B, C, M, K, Ncol, bias, epi);
    };

    // label-conv folded vectors are layer-invariant
    fold_kernel<<<1, 256, 0, stream>>>(lb[0], lb[2], vsrc_lab, CLS);
    fold_kernel<<<1, 256, 0, stream>>>(lb[1], lb[3], vdst_lab, CLS);
    edge_const_kernel<<<1, 64, 0, stream>>>(lb[5], lb[6], c_edge);

    for (int layer = 0; layer < 2; ++layer) {
        const float* xc = layer ? x_buf : x0;
        const float* yc = layer ? y_buf : y0;
        float* x_out = (layer == 1) ? (float*)d_out : x_buf;
        float* y_out = (layer == 1) ? (float*)d_out + N * HID : y_buf;

        // adj-conv folded attention vectors for this layer
        fold_kernel<<<1, 256, 0, stream>>>(cW[layer][0], cW[layer][2], vsrc_adj, IN_C);
        fold_kernel<<<1, 256, 0, stream>>>(cW[layer][1], cW[layer][3], vdst_adj, IN_C);

        // xs = x @ Wsrc [N,256]; ys = y @ Wsrc_lab [N,256]    (WMMA fp32)
        gemm(xc, cW[layer][0], xs, N_NODES, IN_C, NHEAD * HID, nullptr, 0);
        gemm(yc, lb[0],        ys, N_NODES, CLS,  NHEAD * HID, nullptr, 0);

        // per-node logits via folded vectors (avoids x@Wdst GEMM entirely)
        logits_kernel<<<cdiv(N, 256), 256, 0, stream>>>(xc, vsrc_adj, vdst_adj, asa, ada, N_NODES, IN_C);
        logits_kernel<<<cdiv(N, 256), 256, 0, stream>>>(yc, vsrc_lab, vdst_lab, asl, adl, N_NODES, CLS);

        // init reduction buffers
        fill_kernel<<<cdiv(N * 4, 256), 256, 0, stream>>>(amax_a, -INFINITY, (int)(N * 4));
        fill_kernel<<<cdiv(N * 4, 256), 256, 0, stream>>>(den_a, 0.f, (int)(N * 4));
        fill_kernel<<<cdiv(N * 512, 256), 256, 0, stream>>>(agg_a, 0.f, (int)(N * 512));

        // edge sweep 1: alpha + segment max
        edge_alpha_kernel<<<cdiv(E, 256), 256, 0, stream>>>(
            ei, asa, ada, asl, adl, ew + (size_t)layer * E, c_edge,
            alpha_a, alpha_l, amax_a, amax_l);

        // edge sweep 2: fused exp + denom + unnormalized aggregation (both convs)
        edge_agg_kernel<<<(int)E, 64, 0, stream>>>(
            ei, alpha_a, alpha_l, amax_a, amax_l, xs, ys, den_a, den_l, agg_a, agg_l);

        // normalize, head-mean, +bias
        finalize_emb_kernel<<<cdiv(N * HID, 256), 256, 0, stream>>>(
            agg_a, den_a, cW[layer][4], emb_a, agg_l, den_l, lb[4], emb_l);

        // semantic attention: t = tanh(emb @ att_W1 + b1)  (WMMA + fused epilogue)
        gemm(emb_a, att_W1, t_a, N_NODES, HID, ATT_H, att_b1, 1);
        gemm(emb_l, att_W1, t_l, N_NODES, HID, ATT_H, att_b1, 1);
        beta_kernel<<<cdiv(N, 256), 256, 0, stream>>>(t_a, t_l, att_W2, beta);
        combine_kernel<<<cdiv(N * HID, 256), 256, 0, stream>>>(emb_a, emb_l, beta, x_out,
                                                               layer == 0 ? 1 : 0);

        // y = sigmoid(x @ mlp_W + mlp_b)   (WMMA + fused sigmoid epilogue)
        gemm(x_out, mlp_W, y_out, N_NODES, HID, CLS, mlp_b, 2);
    }
}